// Loss4BidirFlowVec_85203561218679
// MI455X (gfx1250) — compile-verified
//
#include <hip/hip_runtime.h>

typedef __attribute__((ext_vector_type(2))) float v2f;
typedef __attribute__((ext_vector_type(8))) float v8f;

#define N_DIM 16
#define H_DIM 768
#define W_DIM 768
#define HW (H_DIM * W_DIM)
#define NHW (N_DIM * HW)
#define EPS_V 0.001f
#define ROWS_PER_BLOCK 8

#define WM1F ((float)(W_DIM - 1))
#define HM1F ((float)(H_DIM - 1))
#define INV_W (1.0f / WM1F)
#define INV_H (1.0f / HM1F)

// Full zero-padded bilinear eval of m1 = warp(GridXY, flo1) at integer pixel
// (cx, cy) given its flow (u, v). Used only on the rare boundary path.
__device__ __forceinline__ void eval_full(float u, float v, float cx, float cy,
                                          float& mx, float& my) {
    const float sx = cx + u;
    const float sy = cy + v;
    const float fx0 = floorf(sx), fy0 = floorf(sy);
    const float fx1 = fx0 + 1.0f, fy1 = fy0 + 1.0f;
    const float ax = sx - fx0, ay = sy - fy0;

    const float vx0 = (fx0 >= 0.0f && fx0 <= WM1F) ? 1.0f : 0.0f;
    const float vx1 = (fx1 >= 0.0f && fx1 <= WM1F) ? 1.0f : 0.0f;
    const float vy0 = (fy0 >= 0.0f && fy0 <= HM1F) ? 1.0f : 0.0f;
    const float vy1 = (fy1 >= 0.0f && fy1 <= HM1F) ? 1.0f : 0.0f;

    const float c00 = (1.0f - ax) * (1.0f - ay) * vx0 * vy0;
    const float c10 = ax * (1.0f - ay) * vx1 * vy0;
    const float c01 = (1.0f - ax) * ay * vx0 * vy1;
    const float c11 = ax * ay * vx1 * vy1;

    const float wsum = c00 + c10 + c01 + c11;
    const float mask = (wsum < 0.9999f) ? 0.0f : 1.0f;

    mx = (fx0 * (c00 + c01) + fx1 * (c10 + c11)) * INV_W * mask;
    my = (fy0 * (c00 + c10) + fy1 * (c01 + c11)) * INV_H * mask;
}

__global__ void __launch_bounds__(256)
cycle_loss_kernel(const float* __restrict__ AtoB,
                  const float* __restrict__ BtoA,
                  float* __restrict__ out) {
    // WEIGHT / (N*H*W): both directions use the identical normalization.
    const float SCALE = 1.0f / (float)NHW;

    // Block -> (dir, n, y0) decomposition done ONCE (uniform, SALU).
    const int gr0 = blockIdx.y * ROWS_PER_BLOCK;
    const int dir = gr0 / (N_DIM * H_DIM);
    const int rem = gr0 - dir * (N_DIM * H_DIM);
    const int n   = rem / H_DIM;
    const int y0  = rem - n * H_DIM;
    const int x   = blockIdx.x * blockDim.x + threadIdx.x;  // [0, 768)

    const float* __restrict__ flo1 = dir ? BtoA : AtoB;
    const float* __restrict__ flo2 = dir ? AtoB : BtoA;
    const int planeBase = n * 2 * HW;       // start of u-plane for image n

    float acc = 0.0f;
    const float gx = (float)x * INV_W;

    int rowBase = planeBase + y0 * W_DIM + x;
    #pragma unroll 2
    for (int ry = 0; ry < ROWS_PER_BLOCK; ++ry, rowBase += W_DIM) {
        const int y = y0 + ry;

        // ---- second warp: sample m1 at (x + u2, y + v2) ----
        const float u2 = flo2[rowBase];
        const float v2 = flo2[rowBase + HW];
        const float sx = (float)x + u2;
        const float sy = (float)y + v2;
        const float fx0 = floorf(sx), fy0 = floorf(sy);
        const float fx1 = fx0 + 1.0f, fy1 = fy0 + 1.0f;
        const float ax = sx - fx0, ay = sy - fy0;

        const float vx0 = (fx0 >= 0.0f && fx0 <= WM1F) ? 1.0f : 0.0f;
        const float vx1 = (fx1 >= 0.0f && fx1 <= WM1F) ? 1.0f : 0.0f;
        const float vy0 = (fy0 >= 0.0f && fy0 <= HM1F) ? 1.0f : 0.0f;
        const float vy1 = (fy1 >= 0.0f && fy1 <= HM1F) ? 1.0f : 0.0f;

        const float c00 = (1.0f - ax) * (1.0f - ay) * vx0 * vy0;
        const float c10 = ax * (1.0f - ay) * vx1 * vy0;
        const float c01 = (1.0f - ax) * ay * vx0 * vy1;
        const float c11 = ax * ay * vx1 * vy1;
        const float wsum = c00 + c10 + c01 + c11;
        const float mask2 = (wsum < 0.9999f) ? 0.0f : 1.0f;

        // Clamped corner coords: always in-bounds (safe loads); invalid outer
        // corners contribute via c==0 so the value there doesn't matter.
        const float cx0 = fminf(fmaxf(fx0, 0.0f), WM1F);
        const float cx1 = fminf(fmaxf(fx1, 0.0f), WM1F);
        const float cy0 = fminf(fmaxf(fy0, 0.0f), HM1F);
        const float cy1 = fminf(fmaxf(fy1, 0.0f), HM1F);
        const int ix0 = (int)cx0, ix1 = (int)cx1;
        const int iy0 = (int)cy0, iy1 = (int)cy1;

        // ---- inner warp, cheap closed form ----
        // m1 at a pixel whose inner sample stays inside the image is exactly
        // ((cx+u)/(W-1), (cy+v)/(H-1)): bilinear interp of a linear grid with
        // total weight 1 and mask 1 (holds on the sx==W-1 edge too).
        const int r0 = planeBase + iy0 * W_DIM;
        const int r1 = planeBase + iy1 * W_DIM;
        const float u00 = flo1[r0 + ix0],      v00 = flo1[r0 + ix0 + HW];
        const float u10 = flo1[r0 + ix1],      v10 = flo1[r0 + ix1 + HW];
        const float u01 = flo1[r1 + ix0],      v01 = flo1[r1 + ix0 + HW];
        const float u11 = flo1[r1 + ix1],      v11 = flo1[r1 + ix1 + HW];

        const float sx00 = cx0 + u00, sy00 = cy0 + v00;
        const float sx10 = cx1 + u10, sy10 = cy0 + v10;
        const float sx01 = cx0 + u01, sy01 = cy1 + v01;
        const float sx11 = cx1 + u11, sy11 = cy1 + v11;

        const bool ez00 = sx00 >= 0.0f && sx00 <= WM1F && sy00 >= 0.0f && sy00 <= HM1F;
        const bool ez10 = sx10 >= 0.0f && sx10 <= WM1F && sy10 >= 0.0f && sy10 <= HM1F;
        const bool ez01 = sx01 >= 0.0f && sx01 <= WM1F && sy01 >= 0.0f && sy01 <= HM1F;
        const bool ez11 = sx11 >= 0.0f && sx11 <= WM1F && sy11 >= 0.0f && sy11 <= HM1F;

        float m00x = sx00 * INV_W, m00y = sy00 * INV_H;
        float m10x = sx10 * INV_W, m10y = sy10 * INV_H;
        float m01x = sx01 * INV_W, m01y = sy01 * INV_H;
        float m11x = sx11 * INV_W, m11y = sy11 * INV_H;

        // Rare boundary path (wave-uniform gate): only waves touching the
        // border band pay for the full zero-padded bilinears.
        if (__builtin_expect(!__all(ez00 && ez10 && ez01 && ez11), 0)) {
            float fx, fy;
            eval_full(u00, v00, cx0, cy0, fx, fy);
            m00x = ez00 ? m00x : fx;  m00y = ez00 ? m00y : fy;
            eval_full(u10, v10, cx1, cy0, fx, fy);
            m10x = ez10 ? m10x : fx;  m10y = ez10 ? m10y : fy;
            eval_full(u01, v01, cx0, cy1, fx, fy);
            m01x = ez01 ? m01x : fx;  m01y = ez01 ? m01y : fy;
            eval_full(u11, v11, cx1, cy1, fx, fy);
            m11x = ez11 ? m11x : fx;  m11y = ez11 ? m11y : fy;
        }

        const float mx = (m00x * c00 + m10x * c10 + m01x * c01 + m11x * c11) * mask2;
        const float my = (m00y * c00 + m10y * c10 + m01y * c01 + m11y * c11) * mask2;

        const float gy = (float)y * INV_H;
        const float dx = gx - mx;
        const float dy = gy - my;
        // Raw v_sqrt_f32 (1 ulp) — no precise-sqrt fixup sequence.
        acc += __builtin_amdgcn_sqrtf(dx * dx + dy * dy + EPS_V * EPS_V);
    }

    // ---- wave32 reduction via f32 WMMA (lossless) ----
    // A (16x4 f32): lanes 0-15 supply rows M=lane at K={0,1}; lanes 16-31
    // supply K={2,3}; B = ones -> D[m][n] = row-sum of A.
    v2f a; a[0] = acc; a[1] = 0.0f;
    v2f b; b[0] = 1.0f; b[1] = 1.0f;
    v8f c = {};
    v8f d = __builtin_amdgcn_wmma_f32_16x16x4_f32(
        /*neg_a=*/false, a, /*neg_b=*/false, b,
        /*c_mod=*/(short)0, c, /*reuse_a=*/false, /*reuse_b=*/false);

    float s = d[0] + d[1] + d[2] + d[3] + d[4] + d[5] + d[6] + d[7];
    // Lanes 0-15 hold sum(rows 0-7), lanes 16-31 hold sum(rows 8-15).
    float t = __shfl_xor(s, 16, 32);
    const float wave_sum = s + t;

    // ---- block reduction in LDS -> exactly ONE atomic per block ----
    __shared__ float smem[8];                 // 256 threads = 8 waves
    const int lane = threadIdx.x & 31;
    const int wid  = threadIdx.x >> 5;
    if (lane == 0) smem[wid] = wave_sum;
    __syncthreads();
    if (threadIdx.x == 0) {
        float blk = smem[0] + smem[1] + smem[2] + smem[3]
                  + smem[4] + smem[5] + smem[6] + smem[7];
        atomicAdd(out, blk * SCALE);
    }
}

extern "C" void kernel_launch(void* const* d_in, const int* in_sizes, int n_in,
                              void* d_out, int out_size, void* d_ws, size_t ws_size,
                              hipStream_t stream) {
    const float* AtoB = (const float*)d_in[0];
    const float* BtoA = (const float*)d_in[1];
    float* out = (float*)d_out;

    // d_out is poisoned by the harness; zero before atomic accumulation.
    hipMemsetAsync(out, 0, sizeof(float) * (size_t)out_size, stream);

    // gridDim.x * 256 covers W exactly (3*256 = 768).
    // gridDim.y covers 2*N*H rows / 8 rows-per-block = 3072.
    dim3 grid(W_DIM / 256, (2 * N_DIM * H_DIM) / ROWS_PER_BLOCK);
    cycle_loss_kernel<<<grid, 256, 0, stream>>>(AtoB, BtoA, out);
}